// CTCLayer_84920093376549
// MI455X (gfx1250) — compile-verified
//
#include <hip/hip_runtime.h>
#include <stdint.h>

// ---------------- problem constants ----------------
#define B_     2048
#define T_     256
#define C_     55
#define L_     32
#define S_     65      // 2*L+1
#define BLANK_ 54
#define NEGF   (-1e30f)

// ---------------- async pipeline constants ----------------
#define DEPTH  24      // prefetch distance (rows in flight; 48 async ops <= 63 cnt limit)
#define RING   25      // ring slots (RING > DEPTH avoids read/write slot clash)
#define NSLOT  26      // ring + 1 dummy slot for tail dummy issues
#define ROWF   56      // floats per slot (55 used, padded)

// ---------------- CDNA5 async global -> LDS helpers ----------------
__device__ __forceinline__ void async_b32(const float* g, float* lds_generic) {
#if defined(__has_builtin)
#if __has_builtin(__builtin_amdgcn_global_load_async_to_lds_b32)
  __builtin_amdgcn_global_load_async_to_lds_b32(
      (__attribute__((address_space(1))) int*)(uintptr_t)g,
      (__attribute__((address_space(3))) int*)(uint32_t)(uintptr_t)lds_generic,
      0, 0);
  return;
#endif
#endif
  unsigned lo = (unsigned)(uintptr_t)lds_generic;   // low 32 bits of flat == LDS offset
  asm volatile("global_load_async_to_lds_b32 %0, %1, off"
               :: "v"(lo), "v"(g) : "memory");
}

// issue one logits row (55 floats) into an LDS slot; lane pattern: c=l and c=32+l
__device__ __forceinline__ void async_row_load(const float* gsrc, float* dst, int l) {
  async_b32(gsrc + l, dst + l);
  if (l < C_ - 32) async_b32(gsrc + 32 + l, dst + 32 + l);
}

template <int N>
__device__ __forceinline__ void wait_async_le() {
#if defined(__has_builtin)
#if __has_builtin(__builtin_amdgcn_s_wait_asynccnt)
  __builtin_amdgcn_s_wait_asynccnt((unsigned short)N);
  asm volatile("" ::: "memory");
  return;
#endif
#endif
  asm volatile("s_wait_asynccnt %0" :: "n"(N) : "memory");
}

// ---------------- math helpers ----------------
__device__ __forceinline__ float lse3(float x, float y, float z) {
  float m = fmaxf(fmaxf(x, y), z);
  return m + __logf(__expf(x - m) + __expf(y - m) + __expf(z - m));
}

// logsumexp over the 55 classes of one row living in LDS; wave32 shfl_xor tree
__device__ __forceinline__ float row_lse(const float* rb, int l) {
  float x0 = rb[l];
  float x1 = (l < C_ - 32) ? rb[32 + l] : NEGF;
  float m = fmaxf(x0, x1);
#pragma unroll
  for (int off = 16; off; off >>= 1) m = fmaxf(m, __shfl_xor(m, off, 32));
  float s = __expf(x0 - m) + __expf(x1 - m);
#pragma unroll
  for (int off = 16; off; off >>= 1) s += __shfl_xor(s, off, 32);
  return m + __logf(s);
}

// ---------------- main CTC kernel: one wave32 per example ----------------
__global__ __launch_bounds__(32) void ctc_kernel(const int* __restrict__ labels,
                                                 const float* __restrict__ logits,
                                                 float* __restrict__ ws) {
  __shared__ float buf[NSLOT * ROWF];
  __shared__ int   slab[L_];
  __shared__ float alph[S_ + 1];

  const int b = blockIdx.x;
  const int l = threadIdx.x;
  const float* base = logits + (size_t)b * (T_ * C_);

  // labels, label length, extended sequence metadata (per-lane states 3l..3l+2)
  int lv = labels[(size_t)b * L_ + l];
  int len = __popcll(__ballot(lv != -1));
  slab[l] = lv < 0 ? 0 : lv;
  __syncthreads();

  int e0, e1, e2, k0, k1, k2;
  {
    int s0 = 3 * l, s1 = 3 * l + 1, s2 = 3 * l + 2;
    // state even -> blank, no skip; state odd -> label[(s-1)/2], skip iff s>=3 && lbl[j]!=lbl[j-1]
    if (s0 & 1) { int j = (s0 - 1) >> 1; e0 = slab[j]; k0 = (s0 >= 3) && (slab[j] != slab[j - 1]); }
    else        { e0 = BLANK_; k0 = 0; }
    if (s1 & 1) { int j = (s1 - 1) >> 1; e1 = slab[j]; k1 = (s1 >= 3) && (slab[j] != slab[j - 1]); }
    else        { e1 = BLANK_; k1 = 0; }
    if (s2 & 1) { int j = (s2 - 1) >> 1; e2 = slab[j]; k2 = (s2 >= 3) && (slab[j] != slab[j - 1]); }
    else        { e2 = BLANK_; k2 = 0; }
  }

  // prologue: fill the async pipeline with rows 0..DEPTH-1 (slots 0..DEPTH-1)
  for (int r = 0; r < DEPTH; r++)
    async_row_load(base + (size_t)r * C_, buf + r * ROWF, l);

  // ring state: rd = slot of current row t; wr = slot for row t+DEPTH
  int rd = 0;
  int wr = DEPTH;                       // (0 + DEPTH) % RING, DEPTH < RING
  const float* gnext = base + (size_t)DEPTH * C_;  // global ptr of row t+DEPTH
  const float* glast = base + (size_t)(T_ - 1) * C_;

  float a0, a1, a2;

  // ---- t = 0 (peeled): init alpha ----
  {
    wait_async_le<2 * (DEPTH - 1)>();
    async_row_load(gnext, buf + wr * ROWF, l);
    gnext += C_;
    const float* rb = buf;              // slot 0
    float lse = row_lse(rb, l);
    a0 = (l == 0) ? rb[BLANK_] - lse : NEGF;
    a1 = (l == 0) ? rb[e1] - lse : NEGF;
    a2 = NEGF;
    rd = 1;
    wr = (wr + 1 == RING) ? 0 : wr + 1;
  }

  // ---- steady-state DP over t = 1..T-1 ----
  for (int t = 1; t < T_; t++) {
    // oldest two outstanding async ops are row t's loads -> wait for them
    wait_async_le<2 * (DEPTH - 1)>();

    // keep pipeline full: issue row t+DEPTH, or a dummy re-read into the spare slot
    if (t + DEPTH < T_) {
      async_row_load(gnext, buf + wr * ROWF, l);
      gnext += C_;
    } else {
      async_row_load(glast, buf + RING * ROWF, l);   // dummy slot, keeps cnt invariant
    }

    const float* rb = buf + rd * ROWF;
    float lse = row_lse(rb, l);
    float lp0 = rb[e0] - lse;
    float lp1 = rb[e1] - lse;
    float lp2 = rb[e2] - lse;
    float p2 = __shfl_up(a2, 1, 32);    // alpha[3l-1] from previous lane
    float p1 = __shfl_up(a1, 1, 32);    // alpha[3l-2] from previous lane
    if (l == 0) { p2 = NEGF; p1 = NEGF; }
    float n0 = lse3(a0, p2, k0 ? p1 : NEGF) + lp0;
    float n1 = lse3(a1, a0, k1 ? p2 : NEGF) + lp1;
    float n2 = lse3(a2, a1, k2 ? a0 : NEGF) + lp2;
    a0 = n0; a1 = n1; a2 = n2;

    rd = (rd + 1 == RING) ? 0 : rd + 1;
    wr = (wr + 1 == RING) ? 0 : wr + 1;
  }

  // final extraction: LSE of alpha_T[2*len] and alpha_T[2*len-1]
  {
    int s = 3 * l;
    if (s     < S_) alph[s]     = a0;
    if (s + 1 < S_) alph[s + 1] = a1;
    if (s + 2 < S_) alph[s + 2] = a2;
  }
  __syncthreads();
  if (l == 0) {
    int idx = 2 * len;
    float xl = alph[idx];
    float xp = alph[idx > 0 ? idx - 1 : 0];
    float m  = fmaxf(xl, xp);
    float ll = m + __logf(__expf(xl - m) + __expf(xp - m));
    ws[b] = -ll;
  }
}

// ---------------- deterministic mean over B values ----------------
__global__ __launch_bounds__(256) void reduce_mean_kernel(const float* __restrict__ ws,
                                                          float* __restrict__ out, int n) {
  __shared__ float sm[256];
  float s = 0.f;
  for (int i = threadIdx.x; i < n; i += 256) s += ws[i];
  sm[threadIdx.x] = s;
  __syncthreads();
  for (int off = 128; off; off >>= 1) {
    if (threadIdx.x < off) sm[threadIdx.x] += sm[threadIdx.x + off];
    __syncthreads();
  }
  if (threadIdx.x == 0) out[0] = sm[0] / (float)n;
}

extern "C" void kernel_launch(void* const* d_in, const int* in_sizes, int n_in,
                              void* d_out, int out_size, void* d_ws, size_t ws_size,
                              hipStream_t stream) {
  const int*   labels = (const int*)d_in[0];    // (B, 32) int32, -1 padded
  const float* logits = (const float*)d_in[1];  // (B, 256, 55) f32
  float* out = (float*)d_out;                   // scalar mean
  float* ws  = (float*)d_ws;                    // per-example -log p (B floats)

  ctc_kernel<<<B_, 32, 0, stream>>>(labels, logits, ws);
  reduce_mean_kernel<<<1, 256, 0, stream>>>(ws, out, B_);
}